// NoisyTopKRouter_21741124452486
// MI455X (gfx1250) — compile-verified
//
#include <hip/hip_runtime.h>
#include <hip/hip_bf16.h>
#include <math.h>

typedef float v2f __attribute__((ext_vector_type(2)));
typedef float v8f __attribute__((ext_vector_type(8)));

#define B_       4
#define S_       4096
#define D_       4096
#define E_       64
#define NTOK     (B_ * S_)
#define TOK_TILE 64                 // M-tile per block (4 M-subtiles per wave)
#define CHUNK_K  128                // K-chunk staged in LDS
#define NCHUNK   (D_ / CHUNK_K)     // 32
#define APAD     4                  // +16B row pad -> 4-bank rotation, no conflicts
#define AROW     (CHUNK_K + APAD)   // 132 floats per LDS row

// --- CDNA5 async global->LDS helpers (CDNA5 ISA) ---
__device__ __forceinline__ uint32_t lds_off_of(const void* p) {
    return (uint32_t)(uintptr_t)p;   // low 32 bits of generic LDS ptr = LDS byte offset
}
__device__ __forceinline__ void async_load_b128(uint32_t lds_byte, const float* gaddr) {
    asm volatile("global_load_async_to_lds_b128 %0, %1, off"
                 :: "v"(lds_byte), "v"(gaddr) : "memory");
}
__device__ __forceinline__ void wait_async_le8() { asm volatile("s_wait_asynccnt 0x8" ::: "memory"); }
__device__ __forceinline__ void wait_async_le0() { asm volatile("s_wait_asynccnt 0x0" ::: "memory"); }

__device__ __forceinline__ bool gt_pair(float v, int i, float ov, int oi) {
    return (v > ov) || (v == ov && i < oi);
}

// ---------------------------------------------------------------------------
// Fused NoisyTopK router: grid.x = NTOK/64, block = 256 (8 wave32 waves).
// Phase 1: dual fp32 GEMM via V_WMMA_F32_16X16X4_F32. 64x128 A tile double-
//          buffered in LDS (async b128), each wave owns one 16-expert N-tile of
//          {router|noise} and 4 M-subtiles sharing the same B registers.
// Phase 2: 64x128 logits tile staged in LDS (+bias).
// Phase 3: softplus-noise combine, wave32 top-2, sparse softmax (8 tok/wave).
// ---------------------------------------------------------------------------
__global__ __launch_bounds__(256)
void fused_router_kernel(const float* __restrict__ x,
                         const float* __restrict__ noise_u,
                         const float* __restrict__ w_router,
                         const float* __restrict__ b_router,
                         const float* __restrict__ w_noise,
                         const float* __restrict__ b_noise,
                         float* __restrict__ probs,    // [NTOK, E]
                         int*   __restrict__ indices)  // [NTOK, 2]
{
    __shared__ float sA[2][TOK_TILE * AROW];     // 2 x 33 KB, padded rows
    __shared__ float sRes[2][TOK_TILE][E_];      // 32 KB: [router|noise][tok][expert]

    const int tid     = threadIdx.x;
    const int lane    = tid & 31;
    const int wave    = tid >> 5;
    const int tokBase = blockIdx.x * TOK_TILE;

    const int  mat  = wave >> 2;                 // 0 = router, 1 = noise
    const int  col0 = (wave & 3) << 4;           // 16-expert N-tile
    const float* W    = mat ? w_noise : w_router;
    const float* bias = mat ? b_noise : b_router;

    const int halfsel = lane >> 4;               // K half split (ISA 7.12.2)
    const int lrow    = lane & 15;
    const int kofs    = halfsel << 1;

    const uint32_t ldsA0 = lds_off_of(&sA[0][0]);
    const uint32_t ldsA1 = lds_off_of(&sA[1][0]);

    // cooperative async copy: 64 rows x 128 f32 = 2048 b128s, 8 per thread
    auto issue_chunk = [&](int c) {
        const uint32_t base = (c & 1) ? ldsA1 : ldsA0;
        const int kBase = c * CHUNK_K;
        #pragma unroll
        for (int j = 0; j < 8; ++j) {
            const int id   = tid + (j << 8);     // 0..2047
            const int row  = id >> 5;            // 64 rows, 32 b128s each
            const int fcol = (id & 31) << 2;     // float column in row
            const float* g = x + (size_t)(tokBase + row) * D_ + kBase + fcol;
            async_load_b128(base + (uint32_t)((row * AROW + fcol) << 2), g);
        }
    };

    issue_chunk(0);

    const float* wcol = W + (size_t)kofs * E_ + (col0 + lrow);
    v8f acc[4];
    #pragma unroll
    for (int s = 0; s < 4; ++s) acc[s] = (v8f){0.f,0.f,0.f,0.f,0.f,0.f,0.f,0.f};

    for (int c = 0; c < NCHUNK; ++c) {
        if (c + 1 < NCHUNK) { issue_chunk(c + 1); wait_async_le8(); }
        else                { wait_async_le0(); }
        __syncthreads();                         // chunk c visible to all waves

        const float* aBuf  = &sA[c & 1][lrow * AROW + kofs];
        const size_t kGlob = (size_t)c * CHUNK_K;
        #pragma unroll 2
        for (int kk = 0; kk < CHUNK_K; kk += 4) {
            v2f b;                               // shared across 4 M-subtiles
            b.x = wcol[(kGlob + kk) * E_];       // W[k+kofs  ][col]
            b.y = wcol[(kGlob + kk + 1) * E_];   // W[k+kofs+1][col]
            #pragma unroll
            for (int s = 0; s < 4; ++s) {
                const v2f a = *(const v2f*)(aBuf + s * (16 * AROW) + kk); // ds_load_b64
                acc[s] = __builtin_amdgcn_wmma_f32_16x16x4_f32(
                    false, a, false, b, (short)0, acc[s], false, false);
            }
        }
        __syncthreads();                         // buffer free for overwrite
    }

    // Phase 2: bias + stage 64x16 tiles into LDS result area
    const float bv = bias[col0 + lrow];
    #pragma unroll
    for (int s = 0; s < 4; ++s) {
        #pragma unroll
        for (int j = 0; j < 8; ++j) {
            const int m = (s << 4) + j + (halfsel << 3);   // C/D layout
            sRes[mat][m][col0 + lrow] = acc[s][j] + bv;
        }
    }
    __syncthreads();

    // Phase 3: epilogue — each wave handles 8 tokens
    #pragma unroll
    for (int t = 0; t < 8; ++t) {
        const int    tl   = (wave << 3) + t;
        const size_t base = (size_t)(tokBase + tl) * E_;

        float nv[2]; int ni[2];
        #pragma unroll
        for (int u = 0; u < 2; ++u) {
            const int e  = lane + (u << 5);
            const float lg = sRes[0][tl][e];
            const float nl = sRes[1][tl][e];
            const float sp = fmaxf(nl, 0.f) + log1pf(expf(-fabsf(nl)));  // stable softplus
            nv[u] = lg + noise_u[base + e] * sp;
            ni[u] = e;
        }

        float v1, v2; int i1, i2;
        if (gt_pair(nv[0], ni[0], nv[1], ni[1])) { v1 = nv[0]; i1 = ni[0]; v2 = nv[1]; i2 = ni[1]; }
        else                                     { v1 = nv[1]; i1 = ni[1]; v2 = nv[0]; i2 = ni[0]; }

        #pragma unroll
        for (int msk = 16; msk >= 1; msk >>= 1) {          // wave32 butterfly top-2
            float o1  = __shfl_xor(v1, msk, 32);
            int   oi1 = __shfl_xor(i1, msk, 32);
            float o2  = __shfl_xor(v2, msk, 32);
            int   oi2 = __shfl_xor(i2, msk, 32);
            if (gt_pair(o1, oi1, v1, i1)) {
                float tv = v1; int ti = i1; v1 = o1; i1 = oi1; o1 = tv; oi1 = ti;
                tv = v2; ti = i2; v2 = o2; i2 = oi2; o2 = tv; oi2 = ti;
            }
            if (gt_pair(o1, oi1, v2, i2)) { v2 = o1; i2 = oi1; }
        }

        const float e2 = expf(v2 - v1);          // only top-2 survive the -inf mask
        const float s  = 1.f + e2;
        const float p1 = 1.f / s;
        const float p2 = e2 / s;

        #pragma unroll
        for (int u = 0; u < 2; ++u) {
            const int e = lane + (u << 5);
            probs[base + e] = (e == i1) ? p1 : ((e == i2) ? p2 : 0.f);
        }
        if (lane == 0) {
            indices[(tokBase + tl) * 2 + 0] = i1;
            indices[(tokBase + tl) * 2 + 1] = i2;
        }
    }
}

// ---------------------------------------------------------------------------
extern "C" void kernel_launch(void* const* d_in, const int* in_sizes, int n_in,
                              void* d_out, int out_size, void* d_ws, size_t ws_size,
                              hipStream_t stream) {
    const float* x        = (const float*)d_in[0];
    const float* noise_u  = (const float*)d_in[1];
    const float* w_router = (const float*)d_in[2];
    const float* b_router = (const float*)d_in[3];
    const float* w_noise  = (const float*)d_in[4];
    const float* b_noise  = (const float*)d_in[5];

    float* probs   = (float*)d_out;                     // [NTOK, E]
    int*   indices = (int*)(probs + (size_t)NTOK * E_); // [NTOK, 2]

    fused_router_kernel<<<NTOK / TOK_TILE, 256, 0, stream>>>(
        x, noise_u, w_router, b_router, w_noise, b_noise, probs, indices);
}